// Net_30872224923828
// MI455X (gfx1250) — compile-verified
//
#include <hip/hip_runtime.h>
#include <math.h>

typedef __attribute__((ext_vector_type(2))) float v2f;
typedef __attribute__((ext_vector_type(8))) float v8f;

#define NEG_SLOPE 0.2f

// ---------------------------------------------------------------------------
// helpers
// ---------------------------------------------------------------------------
static __device__ __forceinline__ void atomicMaxF(float* addr, float val) {
    // monotonic punning trick: works across mixed signs given -inf init
    if (val >= 0.0f) atomicMax((int*)addr, __float_as_int(val));
    else             atomicMin((unsigned int*)addr, __float_as_uint(val));
}

static __device__ __forceinline__ void endpoints(const int* __restrict__ ei,
                                                 int E, int e, int& s, int& d) {
    if (e < E) { s = ei[e]; d = ei[E + e]; }   // edge_index[0][e], edge_index[1][e]
    else       { s = e - E; d = s; }           // appended self-loop
}

// ---------------------------------------------------------------------------
// init: -inf segment-max state, zero segment-sum state, zero output accumulators
// ---------------------------------------------------------------------------
__global__ void init_state(float* __restrict__ m1, float* __restrict__ s1,
                           float* __restrict__ m2, float* __restrict__ s2,
                           float* __restrict__ xemb, float* __restrict__ out2,
                           int N, int H, int HC, int OUT) {
    int t = blockIdx.x * blockDim.x + threadIdx.x;
    const float ninf = __int_as_float(0xFF800000u);
    if (t < N * H)   { m1[t] = ninf; s1[t] = 0.0f; }
    if (t < N)       { m2[t] = ninf; s2[t] = 0.0f; }
    if (t < N * HC)  xemb[t] = 0.0f;
    if (t < N * OUT) out2[t] = 0.0f;
}

// ---------------------------------------------------------------------------
// GEMM1: XL[N,64] = X[N,K] @ W[K,64]  via V_WMMA_F32_16X16X4_F32 (exact fp32)
// one wave per 16-row stripe; 4 column tiles share the A operand
// ---------------------------------------------------------------------------
__global__ void gemm1_wmma(const float* __restrict__ X, const float* __restrict__ W,
                           float* __restrict__ XL, int N, int K) {
    const int lane  = threadIdx.x;        // 0..31
    const int mrow  = lane & 15;          // M index for A, N index for B
    const int khalf = (lane >> 4) << 1;   // lanes 0-15: K=0,1  lanes 16-31: K=2,3
    const int mt    = blockIdx.x;
    int row = mt * 16 + mrow;
    if (row > N - 1) row = N - 1;         // clamp keeps EXEC full for WMMA

    v8f acc0 = {}, acc1 = {}, acc2 = {}, acc3 = {};
    for (int k0 = 0; k0 < K; k0 += 4) {
        const int kb = k0 + khalf;
        v2f a;
        a.x = X[row * K + kb];
        a.y = X[row * K + kb + 1];
        const float* Wr0 = W + kb * 64;
        const float* Wr1 = Wr0 + 64;
        v2f b0, b1, b2, b3;
        b0.x = Wr0[mrow];      b0.y = Wr1[mrow];
        b1.x = Wr0[mrow + 16]; b1.y = Wr1[mrow + 16];
        b2.x = Wr0[mrow + 32]; b2.y = Wr1[mrow + 32];
        b3.x = Wr0[mrow + 48]; b3.y = Wr1[mrow + 48];
        acc0 = __builtin_amdgcn_wmma_f32_16x16x4_f32(false, a, false, b0, (short)0, acc0, false, false);
        acc1 = __builtin_amdgcn_wmma_f32_16x16x4_f32(false, a, false, b1, (short)0, acc1, false, false);
        acc2 = __builtin_amdgcn_wmma_f32_16x16x4_f32(false, a, false, b2, (short)0, acc2, false, false);
        acc3 = __builtin_amdgcn_wmma_f32_16x16x4_f32(false, a, false, b3, (short)0, acc3, false, false);
    }
    const int r0 = (lane >> 4) * 8;       // C/D layout: lanes 16-31 hold M=8..15
    for (int v = 0; v < 8; ++v) {
        int r = mt * 16 + r0 + v;
        if (r < N) {
            float* o = XL + r * 64;
            o[mrow]      = acc0[v];
            o[mrow + 16] = acc1[v];
            o[mrow + 32] = acc2[v];
            o[mrow + 48] = acc3[v];
        }
    }
}

// ---------------------------------------------------------------------------
// GEMM2: XL[N,OUT] = H[N,K] @ W[K,OUT], OUT<=16, B columns >= OUT are zero
// ---------------------------------------------------------------------------
__global__ void gemm2_wmma(const float* __restrict__ Hh, const float* __restrict__ W,
                           float* __restrict__ XL, int N, int K, int OUT) {
    const int lane  = threadIdx.x;
    const int mrow  = lane & 15;
    const int khalf = (lane >> 4) << 1;
    const int mt    = blockIdx.x;
    int row = mt * 16 + mrow;
    if (row > N - 1) row = N - 1;

    v8f acc = {};
    for (int k0 = 0; k0 < K; k0 += 4) {
        const int kb = k0 + khalf;
        v2f a, b;
        a.x = Hh[row * K + kb];
        a.y = Hh[row * K + kb + 1];
        b.x = (mrow < OUT) ? W[kb * OUT + mrow]       : 0.0f;   // select, EXEC stays full
        b.y = (mrow < OUT) ? W[(kb + 1) * OUT + mrow] : 0.0f;
        acc = __builtin_amdgcn_wmma_f32_16x16x4_f32(false, a, false, b, (short)0, acc, false, false);
    }
    if (mrow < OUT) {
        const int r0 = (lane >> 4) * 8;
        for (int v = 0; v < 8; ++v) {
            int r = mt * 16 + r0 + v;
            if (r < N) XL[r * OUT + mrow] = acc[v];
        }
    }
}

// ---------------------------------------------------------------------------
// per-node attention coefficients: alpha_s/alpha_d = <xl[n,h,:], a[h,:]>
// ---------------------------------------------------------------------------
template <int H, int C>
__global__ void coefs(const float* __restrict__ XL, const float* __restrict__ Asrc,
                      const float* __restrict__ Adst, float* __restrict__ as_o,
                      float* __restrict__ ad_o, int N) {
    int t = blockIdx.x * blockDim.x + threadIdx.x;   // t = n*H + h
    if (t >= N * H) return;
    int h = t % H;
    const float* v = XL + t * C;
    float ss = 0.f, sd = 0.f;
#pragma unroll
    for (int c = 0; c < C; ++c) { ss += v[c] * Asrc[h * C + c]; sd += v[c] * Adst[h * C + c]; }
    as_o[t] = ss; ad_o[t] = sd;
}

// ---------------------------------------------------------------------------
// softmax pass a: e = leaky_relu(as[src]+ad[dst]); segment max into m[dst]
// ---------------------------------------------------------------------------
template <int H>
__global__ void edge_lrelu_max(const int* __restrict__ ei, const float* __restrict__ as,
                               const float* __restrict__ ad, float* __restrict__ ebuf,
                               float* __restrict__ m, int E, int Ep) {
    int t = blockIdx.x * blockDim.x + threadIdx.x;
    if (t >= Ep * H) return;
    int e = t / H, h = t - e * H;
    int s, d; endpoints(ei, E, e, s, d);
    float v = as[s * H + h] + ad[d * H + h];
    v = (v > 0.f) ? v : NEG_SLOPE * v;
    ebuf[t] = v;
    atomicMaxF(&m[d * H + h], v);
}

// ---------------------------------------------------------------------------
// softmax pass b: e = exp(e - m[dst]); segment sum into s[dst]
// ---------------------------------------------------------------------------
template <int H>
__global__ void edge_exp_sum(const int* __restrict__ ei, float* __restrict__ ebuf,
                             const float* __restrict__ m, float* __restrict__ ssum,
                             int E, int Ep) {
    int t = blockIdx.x * blockDim.x + threadIdx.x;
    if (t >= Ep * H) return;
    int e = t / H, h = t - e * H;
    int s, d; endpoints(ei, E, e, s, d);
    float v = expf(ebuf[t] - m[d * H + h]);
    ebuf[t] = v;
    atomicAdd(&ssum[d * H + h], v);
}

// ---------------------------------------------------------------------------
// softmax pass c + aggregation: acc[dst,:] += xl[src,:] * (e / (s[dst]+eps))
// one thread per (edge, channel) -> coalesced gathers, L2-resident fp32 atomics
// ---------------------------------------------------------------------------
template <int H, int C>
__global__ void edge_msg(const int* __restrict__ ei, const float* __restrict__ XL,
                         const float* __restrict__ ebuf, const float* __restrict__ ssum,
                         float* __restrict__ acc, int E, int Ep) {
    int t = blockIdx.x * blockDim.x + threadIdx.x;
    if (t >= Ep * (H * C)) return;
    int e  = t / (H * C);
    int hc = t - e * (H * C);
    int h  = hc / C;
    int s, d; endpoints(ei, E, e, s, d);
    float alpha = ebuf[e * H + h] / (ssum[d * H + h] + 1e-16f);
    atomicAdd(&acc[d * (H * C) + hc], XL[s * (H * C) + hc] * alpha);
}

// ---------------------------------------------------------------------------
// finalize layer 1: x_emb = acc + b1 (output #2), h = relu(x_emb)
// ---------------------------------------------------------------------------
__global__ void finalize1(float* __restrict__ xemb, const float* __restrict__ b,
                          float* __restrict__ Hh, int N, int F) {
    int t = blockIdx.x * blockDim.x + threadIdx.x;
    if (t >= N * F) return;
    float v = xemb[t] + b[t % F];
    xemb[t] = v;
    Hh[t] = v > 0.f ? v : 0.f;
}

// finalize layer 2: out = acc + b2 (heads=1, mean over 1 head == identity)
__global__ void finalize2(float* __restrict__ out, const float* __restrict__ b,
                          int N, int F) {
    int t = blockIdx.x * blockDim.x + threadIdx.x;
    if (t >= N * F) return;
    out[t] += b[t % F];
}

// ---------------------------------------------------------------------------
// launch
// ---------------------------------------------------------------------------
extern "C" void kernel_launch(void* const* d_in, const int* in_sizes, int n_in,
                              void* d_out, int out_size, void* d_ws, size_t ws_size,
                              hipStream_t stream) {
    const float* x    = (const float*)d_in[0];
    const int*   ei   = (const int*)d_in[1];
    const float* W1   = (const float*)d_in[2];
    const float* as1w = (const float*)d_in[3];
    const float* ad1w = (const float*)d_in[4];
    const float* b1   = (const float*)d_in[5];
    const float* W2   = (const float*)d_in[6];
    const float* as2w = (const float*)d_in[7];
    const float* ad2w = (const float*)d_in[8];
    const float* b2   = (const float*)d_in[9];

    const int HC  = in_sizes[5];        // HEADS*HID = 64
    const int IN  = in_sizes[2] / HC;   // 128
    const int OUT = in_sizes[9];        // 8
    const int N   = in_sizes[0] / IN;   // 100000
    const int E   = in_sizes[1] / 2;    // 1.6M
    const int Ep  = E + N;              // + self loops
    const int H   = 4;                  // layer-1 heads (HC == H*16)

    float* out  = (float*)d_out;                 // output #1: [N, OUT]
    float* xemb = out + (size_t)N * OUT;         // output #2: [N, HC]

    float* ws   = (float*)d_ws;
    float* xl1  = ws; ws += (size_t)N * HC;      // layer-1 projected features
    float* as1  = ws; ws += (size_t)N * H;
    float* ad1  = ws; ws += (size_t)N * H;
    float* m1   = ws; ws += (size_t)N * H;
    float* s1   = ws; ws += (size_t)N * H;
    float* e1   = ws; ws += (size_t)Ep * H;
    float* hbuf = ws; ws += (size_t)N * HC;      // relu(x_emb)
    float* xl2  = ws; ws += (size_t)N * OUT;
    float* as2  = ws; ws += (size_t)N;
    float* ad2  = ws; ws += (size_t)N;
    float* m2   = ws; ws += (size_t)N;
    float* s2   = ws; ws += (size_t)N;
    float* e2   = ws; ws += (size_t)Ep;

    const int B = 256;
    auto g = [](int n, int b) { return (n + b - 1) / b; };

    init_state<<<g(N * HC, B), B, 0, stream>>>(m1, s1, m2, s2, xemb, out, N, H, HC, OUT);

    // ----- layer 1 -----
    gemm1_wmma<<<g(N, 16), 32, 0, stream>>>(x, W1, xl1, N, IN);
    coefs<4, 16><<<g(N * H, B), B, 0, stream>>>(xl1, as1w, ad1w, as1, ad1, N);
    edge_lrelu_max<4><<<g(Ep * H, B), B, 0, stream>>>(ei, as1, ad1, e1, m1, E, Ep);
    edge_exp_sum<4><<<g(Ep * H, B), B, 0, stream>>>(ei, e1, m1, s1, E, Ep);
    edge_msg<4, 16><<<g(Ep * H * 16, B), B, 0, stream>>>(ei, xl1, e1, s1, xemb, E, Ep);
    finalize1<<<g(N * HC, B), B, 0, stream>>>(xemb, b1, hbuf, N, HC);

    // ----- layer 2 -----
    gemm2_wmma<<<g(N, 16), 32, 0, stream>>>(hbuf, W2, xl2, N, HC, OUT);
    coefs<1, 8><<<g(N, B), B, 0, stream>>>(xl2, as2w, ad2w, as2, ad2, N);
    edge_lrelu_max<1><<<g(Ep, B), B, 0, stream>>>(ei, as2, ad2, e2, m2, E, Ep);
    edge_exp_sum<1><<<g(Ep, B), B, 0, stream>>>(ei, e2, m2, s2, E, Ep);
    edge_msg<1, 8><<<g(Ep * OUT, B), B, 0, stream>>>(ei, xl2, e2, s2, out, E, Ep);
    finalize2<<<g(N * OUT, B), B, 0, stream>>>(out, b2, N, OUT);
}